// EDcoder_18348100289074
// MI455X (gfx1250) — compile-verified
//
#include <hip/hip_runtime.h>
#include <cstddef>

#define D 128
#define BN_EPS 1e-5f

typedef __attribute__((ext_vector_type(2))) float v2f;
typedef __attribute__((ext_vector_type(8))) float v8f;

// ---------------- small utility kernels ----------------

__global__ void k_zero_u32(unsigned* __restrict__ p, int n) {
    int t = blockIdx.x * blockDim.x + threadIdx.x;
    if (t < n) p[t] = 0u;
}

__global__ void k_zero_f32(float* __restrict__ p, int n) {
    int t = blockIdx.x * blockDim.x + threadIdx.x;
    if (t < n) p[t] = 0.0f;
}

__global__ void k_copy_f32(const float* __restrict__ src, float* __restrict__ dst, size_t n) {
    size_t t = (size_t)blockIdx.x * blockDim.x + threadIdx.x;
    if (t < n) dst[t] = src[t];
}

// degree of A+I based on destinations
__global__ void k_degree(const int* __restrict__ dst, unsigned* __restrict__ deg, int E) {
    int t = blockIdx.x * blockDim.x + threadIdx.x;
    if (t < E) atomicAdd(&deg[dst[t]], 1u);
}

__global__ void k_dinv(const unsigned* __restrict__ deg, float* __restrict__ dinv, int n) {
    int t = blockIdx.x * blockDim.x + threadIdx.x;
    if (t < n) dinv[t] = rsqrtf((float)deg[t] + 1.0f);
}

// ---------------- WMMA GEMM: hw = h @ W ; agg = hw*dinv^2 + bias ----------------
// 256 threads = 8 waves (wave32). Block computes a 16-row x 128-col tile.
// Wave w computes the 16x16 output tile at columns [16w,16w+16).
// Uses V_WMMA_F32_16X16X4_F32 with K stepped by 4; W staged in LDS (64 KB).
__global__ __launch_bounds__(256)
void k_gemm_selfloop(const float* __restrict__ h, const float* __restrict__ Wl,
                     const float* __restrict__ bl, const float* __restrict__ dinv,
                     float* __restrict__ hw, float* __restrict__ agg, int Nn) {
    __shared__ float Ws[D * D];  // 64 KB of the 320 KB WGP LDS
#pragma unroll
    for (int i = 0; i < (D * D) / 256; ++i)
        Ws[i * 256 + threadIdx.x] = Wl[i * 256 + threadIdx.x];
    __syncthreads();

    const int wave = threadIdx.x >> 5;
    const int lane = threadIdx.x & 31;
    const int half = lane >> 4;        // K-half selector (ISA A/B layout)
    const int r16  = lane & 15;        // row within tile (A) / col within tile (B)
    const int colBase = wave * 16;

    int rowA = blockIdx.x * 16 + r16;
    if (rowA >= Nn) rowA = Nn - 1;     // clamp loads, keep EXEC all-ones for WMMA
    const float* hrow = h + (size_t)rowA * D;

    v8f acc = {};
#pragma unroll 8
    for (int kk = 0; kk < D; kk += 4) {
        const int ka = kk + half * 2;
        // A: 16x4 fp32 tile, VGPR0 = {K=0 | K=2}, VGPR1 = {K=1 | K=3}
        const float2 av = *(const float2*)(hrow + ka);
        v2f a; a.x = av.x; a.y = av.y;
        // B: 4x16 fp32 tile, same K-half split across lane groups
        v2f b;
        b.x = Ws[ka * D + colBase + r16];
        b.y = Ws[(ka + 1) * D + colBase + r16];
        acc = __builtin_amdgcn_wmma_f32_16x16x4_f32(
            /*neg_a=*/false, a, /*neg_b=*/false, b,
            /*c_mod=*/(short)0, acc, /*reuse_a=*/false, /*reuse_b=*/false);
    }

    // D layout: VGPR i -> M = i + 8*half, N = lane&15
    const int col = colBase + r16;
    const float bias = bl[col];
#pragma unroll
    for (int i = 0; i < 8; ++i) {
        const int row = blockIdx.x * 16 + half * 8 + i;
        if (row < Nn) {
            const float v  = acc[i];
            const float di = dinv[row];
            hw[(size_t)row * D + col]  = v;
            agg[(size_t)row * D + col] = v * di * di + bias;  // self-loop + bias
        }
    }
}

// ---------------- edge scatter: agg[dst] += hw[src] * dinv[src]*dinv[dst] ----------------
// 32 lanes per edge, float4 per lane -> one coalesced 512B row gather per edge.
__global__ __launch_bounds__(256)
void k_edge_scatter(const int* __restrict__ src, const int* __restrict__ dst,
                    const float* __restrict__ dinv, const float* __restrict__ hw,
                    float* __restrict__ agg, int E) {
    const int t = blockIdx.x * blockDim.x + threadIdx.x;
    const int e = t >> 5;
    if (e >= E) return;
    const int lane = t & 31;
    const int s = src[e];
    const int d = dst[e];
    const float coef = dinv[s] * dinv[d];
    const int c0 = lane * 4;
    const float4 v = *(const float4*)(hw + (size_t)s * D + c0);
    float* a = agg + (size_t)d * D + c0;
    __hip_atomic_fetch_add(a + 0, v.x * coef, __ATOMIC_RELAXED, __HIP_MEMORY_SCOPE_AGENT);
    __hip_atomic_fetch_add(a + 1, v.y * coef, __ATOMIC_RELAXED, __HIP_MEMORY_SCOPE_AGENT);
    __hip_atomic_fetch_add(a + 2, v.z * coef, __ATOMIC_RELAXED, __HIP_MEMORY_SCOPE_AGENT);
    __hip_atomic_fetch_add(a + 3, v.w * coef, __ATOMIC_RELAXED, __HIP_MEMORY_SCOPE_AGENT);
}

// ---------------- ReLU in-place + per-channel sum/sumsq partials ----------------
// 128 threads (one per channel), 64 rows per block, 2 atomics per channel per block.
__global__ __launch_bounds__(128)
void k_relu_stats(float* __restrict__ agg, float* __restrict__ stats, int Nn) {
    const int c  = threadIdx.x;
    const int r0 = blockIdx.x * 64;
    const int r1 = min(r0 + 64, Nn);
    float s = 0.0f, s2 = 0.0f;
    for (int r = r0; r < r1; ++r) {
        const size_t idx = (size_t)r * D + c;
        float v = agg[idx];
        v = fmaxf(v, 0.0f);
        agg[idx] = v;
        s  += v;
        s2 += v * v;
    }
    __hip_atomic_fetch_add(&stats[c],     s,  __ATOMIC_RELAXED, __HIP_MEMORY_SCOPE_AGENT);
    __hip_atomic_fetch_add(&stats[D + c], s2, __ATOMIC_RELAXED, __HIP_MEMORY_SCOPE_AGENT);
}

// ---------------- fold BN stats into per-channel scale/shift ----------------
__global__ void k_bn_finish(float* __restrict__ stats, const float* __restrict__ g,
                            const float* __restrict__ bt, int Nn) {
    const int c = threadIdx.x;
    const float inv_n = 1.0f / (float)Nn;
    const float mean = stats[c] * inv_n;
    const float var  = stats[D + c] * inv_n - mean * mean;   // biased variance
    const float sc   = g[c] * rsqrtf(var + BN_EPS);
    stats[2 * D + c] = sc;
    stats[3 * D + c] = bt[c] - mean * sc;
}

// ---------------- h = agg*scale+shift ; out += h ----------------
__global__ __launch_bounds__(256)
void k_apply_bn(const float* __restrict__ agg, const float* __restrict__ stats,
                float* __restrict__ hnext, float* __restrict__ out, size_t n) {
    const size_t t = (size_t)blockIdx.x * blockDim.x + threadIdx.x;
    if (t >= n) return;
    const int c = (int)(t & (D - 1));
    const float v = agg[t] * stats[2 * D + c] + stats[3 * D + c];
    hnext[t] = v;
    out[t] += v;
}

// ---------------- last layer: out += relu(agg) ----------------
__global__ __launch_bounds__(256)
void k_apply_last(const float* __restrict__ agg, float* __restrict__ out, size_t n) {
    const size_t t = (size_t)blockIdx.x * blockDim.x + threadIdx.x;
    if (t >= n) return;
    out[t] += fmaxf(agg[t], 0.0f);
}

// ---------------- launcher ----------------
extern "C" void kernel_launch(void* const* d_in, const int* in_sizes, int n_in,
                              void* d_out, int out_size, void* d_ws, size_t ws_size,
                              hipStream_t stream) {
    const float* x     = (const float*)d_in[0];
    const int*   ei    = (const int*)  d_in[1];
    const float* W     = (const float*)d_in[2];
    const float* b     = (const float*)d_in[3];
    const float* gamma = (const float*)d_in[4];
    const float* beta  = (const float*)d_in[5];
    float* out = (float*)d_out;

    const int Nn = in_sizes[0] / D;
    const int Ee = in_sizes[1] / 2;
    const size_t ND = (size_t)Nn * D;

    char* ws = (char*)d_ws;
    float*    hw    = (float*)(ws);
    float*    agg   = (float*)(ws + ND * 4);
    float*    hA    = (float*)(ws + 2 * ND * 4);
    float*    hB    = (float*)(ws + 3 * ND * 4);
    unsigned* deg   = (unsigned*)(ws + 4 * ND * 4);
    float*    dinv  = (float*)(ws + 4 * ND * 4 + (size_t)Nn * 4);
    float*    stats = (float*)(ws + 4 * ND * 4 + 2 * (size_t)Nn * 4);

    const int* srcv = ei;
    const int* dstv = ei + Ee;

    // degrees + symmetric norm (recomputed every call: no cross-call state)
    k_zero_u32<<<(Nn + 255) / 256, 256, 0, stream>>>(deg, Nn);
    k_degree  <<<(Ee + 255) / 256, 256, 0, stream>>>(dstv, deg, Ee);
    k_dinv    <<<(Nn + 255) / 256, 256, 0, stream>>>(deg, dinv, Nn);

    // JK='sum' starts from the input
    k_copy_f32<<<(int)((ND + 255) / 256), 256, 0, stream>>>(x, out, ND);

    const float* hcur = x;
    for (int l = 0; l < 3; ++l) {
        k_gemm_selfloop<<<(Nn + 15) / 16, 256, 0, stream>>>(
            hcur, W + (size_t)l * D * D, b + (size_t)l * D, dinv, hw, agg, Nn);
        k_edge_scatter<<<(Ee + 7) / 8, 256, 0, stream>>>(srcv, dstv, dinv, hw, agg, Ee);
        if (l < 2) {
            k_zero_f32 <<<1, 256, 0, stream>>>(stats, 2 * D);
            k_relu_stats<<<(Nn + 63) / 64, D, 0, stream>>>(agg, stats, Nn);
            k_bn_finish <<<1, D, 0, stream>>>(stats, gamma + (size_t)l * D,
                                              beta + (size_t)l * D, Nn);
            float* hn = (l == 0) ? hA : hB;
            k_apply_bn<<<(int)((ND + 255) / 256), 256, 0, stream>>>(agg, stats, hn, out, ND);
            hcur = hn;
        } else {
            k_apply_last<<<(int)((ND + 255) / 256), 256, 0, stream>>>(agg, out, ND);
        }
    }
}